// FCNetwork3D_51573967290604
// MI455X (gfx1250) — compile-verified
//
#include <hip/hip_runtime.h>
#include <hip/hip_bf16.h>
#include <math.h>

#define OBS 2264
#define ACT 32
#define HID 4096
#define VOX 216
#define PRE (OBS - VOX)   // 2048

typedef float v2f __attribute__((ext_vector_type(2)));
typedef float v8f __attribute__((ext_vector_type(8)));

// ---------------------------------------------------------------------------
// Async global->LDS b128 copy (CDNA5 GLOBAL_LOAD_ASYNC_TO_LDS_B128, ASYNCcnt).
// Inline asm per the bridge doc (portable across toolchains; builtin nominal
// pointer types vary). vdst = LDS byte offset VGPR, vaddr = 64-bit global
// address VGPR pair. LDS offset = low 32 bits of the flat shared-pointer
// address.
// ---------------------------------------------------------------------------
__device__ __forceinline__ void async_copy_b128(const float* g, float* l) {
    const uint32_t loff = (uint32_t)(uintptr_t)l;
    asm volatile("global_load_async_to_lds_b128 %0, %1, off"
                 :: "v"(loff), "v"(g) : "memory");
}

__device__ __forceinline__ void wait_async_all() {
    asm volatile("s_wait_asynccnt 0" ::: "memory");
}

// ---------------------------------------------------------------------------
// Kernel 1: conv3d chain (6 layers, SAME, relu) on the 216-voxel tail + input
// normalization. Single block; writes normalized t[OBS] into workspace.
// ---------------------------------------------------------------------------
__global__ __launch_bounds__(256) void prep_kernel(
    const float* __restrict__ x,
    const float* __restrict__ cw0, const float* __restrict__ cb0,
    const float* __restrict__ cw1, const float* __restrict__ cb1,
    const float* __restrict__ cw2, const float* __restrict__ cb2,
    const float* __restrict__ cw3, const float* __restrict__ cb3,
    const float* __restrict__ cw4, const float* __restrict__ cb4,
    const float* __restrict__ cw5, const float* __restrict__ cb5,
    const float* __restrict__ in_shift,
    const float* __restrict__ in_scale,
    float* __restrict__ t)
{
    __shared__ float va[VOX];
    __shared__ float vb[VOX];
    const float* cws[6] = {cw0, cw1, cw2, cw3, cw4, cw5};
    const float* cbs[6] = {cb0, cb1, cb2, cb3, cb4, cb5};

    const int tid = threadIdx.x;
    if (tid < VOX) va[tid] = x[PRE + tid];
    __syncthreads();

    float* src = va;
    float* dst = vb;
    for (int l = 0; l < 6; ++l) {
        float kv[27];
        #pragma unroll
        for (int j = 0; j < 27; ++j) kv[j] = cws[l][j];
        const float bias = cbs[l][0];

        float r = 0.0f;
        if (tid < VOX) {
            const int d = tid / 36;
            const int h = (tid / 6) % 6;
            const int w = tid % 6;
            float s = 0.0f;
            #pragma unroll
            for (int kd = 0; kd < 3; ++kd)
              #pragma unroll
              for (int kh = 0; kh < 3; ++kh)
                #pragma unroll
                for (int kw = 0; kw < 3; ++kw) {
                    const int dd = d + kd - 1;
                    const int hh = h + kh - 1;
                    const int ww = w + kw - 1;
                    if (dd >= 0 && dd < 6 && hh >= 0 && hh < 6 && ww >= 0 && ww < 6)
                        s += src[(dd * 6 + hh) * 6 + ww] * kv[kd * 9 + kh * 3 + kw];
                }
            s += bias;
            r = s > 0.0f ? s : 0.0f;
        }
        if (tid < VOX) dst[tid] = r;
        __syncthreads();
        float* tmp = src; src = dst; dst = tmp;
    }
    // after 6 swaps the final result is back in va (== src)
    for (int i = tid; i < OBS; i += blockDim.x) {
        const float raw = (i < PRE) ? x[i] : src[i - PRE];
        t[i] = (raw - in_shift[i]) / (in_scale[i] + 1e-8f);
    }
}

// ---------------------------------------------------------------------------
// WMMA fp32 GEMV: y[rb..rb+15] = act(W[rb..rb+15][:] @ vin + bias)
// Block = 256 threads = 8 wave32s. The input vector is staged to LDS once per
// block with GLOBAL_LOAD_ASYNC_TO_LDS_B128; the hot loop is then 4 b64 global
// A-loads + 4 ds_load b64 B-broadcasts + 4 V_WMMA_F32_16X16X4_F32 per
// iteration (two K-chunks of 8 per wave per iteration, 8-wave K split).
// A b64 load at W+row*K+k+2*half maps exactly onto the ISA 16x4 fp32 A layout;
// B rows are constant across N, so every D column holds the partial dot.
// MODE 0: tanh.  MODE 1: affine output (scale/shift).
// ---------------------------------------------------------------------------
template <int K, int MODE>
__global__ __launch_bounds__(256) void gemv_wmma(
    const float* __restrict__ W,
    const float* __restrict__ bias,
    const float* __restrict__ vin,
    const float* __restrict__ scale,
    const float* __restrict__ shift,
    float* __restrict__ vout)
{
    constexpr int NW  = 8;        // waves per block
    constexpr int NCH = K / 8;    // K-chunks of 8 (2264 and 4096 both divide)

    __shared__ float lds_vin[K];
    __shared__ float part[NW][16];

    const int tid  = threadIdx.x;
    const int w    = tid >> 5;    // wave id 0..7
    const int lane = tid & 31;
    const int half = lane >> 4;   // 0: K=0,1  1: K=2,3 (ISA A 16x4 fp32 layout)
    const int m    = lane & 15;   // output row within the 16-row tile
    const int rb   = blockIdx.x << 4;

    // --- stage the input vector into LDS via async-to-LDS (ASYNCcnt) ---
    for (int i = tid * 4; i < K; i += 256 * 4)
        async_copy_b128(vin + i, &lds_vin[i]);
    wait_async_all();
    __syncthreads();

    const float* __restrict__ Wrow = W + (size_t)(rb + m) * K;

    v8f c = {};
    for (int ch = 2 * w; ch < NCH; ch += 2 * NW) {
        {
            const int ko = ch * 8 + 2 * half;
            __builtin_prefetch(Wrow + ko + 2 * NW * 8, 0, 3);
            const v2f a0 = *reinterpret_cast<const v2f*>(Wrow + ko);
            const v2f a1 = *reinterpret_cast<const v2f*>(Wrow + ko + 4);
            const v2f b0 = *reinterpret_cast<const v2f*>(lds_vin + ko);
            const v2f b1 = *reinterpret_cast<const v2f*>(lds_vin + ko + 4);
            c = __builtin_amdgcn_wmma_f32_16x16x4_f32(false, a0, false, b0,
                                                      (short)0, c, false, false);
            c = __builtin_amdgcn_wmma_f32_16x16x4_f32(false, a1, false, b1,
                                                      (short)0, c, false, false);
        }
        if (ch + 1 < NCH) {               // wave-uniform guard (NCH may be odd)
            const int ko = (ch + 1) * 8 + 2 * half;
            const v2f a0 = *reinterpret_cast<const v2f*>(Wrow + ko);
            const v2f a1 = *reinterpret_cast<const v2f*>(Wrow + ko + 4);
            const v2f b0 = *reinterpret_cast<const v2f*>(lds_vin + ko);
            const v2f b1 = *reinterpret_cast<const v2f*>(lds_vin + ko + 4);
            c = __builtin_amdgcn_wmma_f32_16x16x4_f32(false, a0, false, b0,
                                                      (short)0, c, false, false);
            c = __builtin_amdgcn_wmma_f32_16x16x4_f32(false, a1, false, b1,
                                                      (short)0, c, false, false);
        }
    }

    // Lane 0 holds rows rb+0..7 (N=0 column), lane 16 holds rows rb+8..15.
    if (m == 0) {
        #pragma unroll
        for (int r = 0; r < 8; ++r) part[w][half * 8 + r] = c[r];
    }
    __syncthreads();

    if (tid < 16) {
        float acc = 0.0f;
        #pragma unroll
        for (int i = 0; i < NW; ++i) acc += part[i][tid];
        acc += bias[rb + tid];
        if (MODE == 0) {
            acc = tanhf(acc);
        } else {
            acc = acc * scale[rb + tid] + shift[rb + tid];
        }
        vout[rb + tid] = acc;
    }
}

// ---------------------------------------------------------------------------
// Launch: prep -> GEMV(W0)+tanh -> GEMV(W1)+tanh -> GEMV(W2)+out affine.
// Stream ordering provides the dependencies (graph-capture safe).
// ---------------------------------------------------------------------------
extern "C" void kernel_launch(void* const* d_in, const int* in_sizes, int n_in,
                              void* d_out, int out_size, void* d_ws, size_t ws_size,
                              hipStream_t stream)
{
    (void)in_sizes; (void)n_in; (void)out_size; (void)ws_size;

    const float* x = (const float*)d_in[0];
    // dict order: x, (cw0,cb0) ... (cw5,cb5), W0,b0,W1,b1,W2,b2,
    //             in_shift, in_scale, out_shift, out_scale
    const float* cw[6];
    const float* cb[6];
    for (int i = 0; i < 6; ++i) {
        cw[i] = (const float*)d_in[1 + 2 * i];
        cb[i] = (const float*)d_in[2 + 2 * i];
    }
    const float* W0        = (const float*)d_in[13];
    const float* b0        = (const float*)d_in[14];
    const float* W1        = (const float*)d_in[15];
    const float* b1        = (const float*)d_in[16];
    const float* W2        = (const float*)d_in[17];
    const float* b2        = (const float*)d_in[18];
    const float* in_shift  = (const float*)d_in[19];
    const float* in_scale  = (const float*)d_in[20];
    const float* out_shift = (const float*)d_in[21];
    const float* out_scale = (const float*)d_in[22];

    float* ws = (float*)d_ws;
    float* t  = ws;              // [OBS]
    float* h0 = t + OBS;         // [HID]
    float* h1 = h0 + HID;        // [HID]
    float* out = (float*)d_out;  // [ACT]

    prep_kernel<<<1, 256, 0, stream>>>(x,
        cw[0], cb[0], cw[1], cb[1], cw[2], cb[2],
        cw[3], cb[3], cw[4], cb[4], cw[5], cb[5],
        in_shift, in_scale, t);

    gemv_wmma<OBS, 0><<<HID / 16, 256, 0, stream>>>(W0, b0, t,  nullptr,   nullptr,   h0);
    gemv_wmma<HID, 0><<<HID / 16, 256, 0, stream>>>(W1, b1, h0, nullptr,   nullptr,   h1);
    gemv_wmma<HID, 1><<<ACT / 16, 256, 0, stream>>>(W2, b2, h1, out_scale, out_shift, out);
}